// GCBlock_61993557950615
// MI455X (gfx1250) — compile-verified
//
#include <hip/hip_runtime.h>
#include <hip/hip_bf16.h>

typedef float v2f __attribute__((ext_vector_type(2)));
typedef float v8f __attribute__((ext_vector_type(8)));

#define N_ATOMS_C 25000
#define N_PAIRS_C 500000
#define N_TILES   (N_PAIRS_C / 16)

// LDS layout (floats)
#define WPI_OFF   0          // 128*256 paired -> 32768
#define WII_OFF   32768      // 32*128 paired  -> 4096
#define WIX_OFF   36864      // 64*64 paired   -> 4096
#define WJX_OFF   40960      // 64*64 paired   -> 4096
#define I1P_OFF   45056      // 8 waves * 16*32 = 4096
#define LDS_FLOATS 49152
#define LDS_BYTES (LDS_FLOATS * 4)   // 192 KB of the 320 KB WGP LDS

static __device__ __forceinline__ v8f wmma4(v2f a, v2f b, v8f c) {
    // V_WMMA_F32_16X16X4_F32 : D = A(16x4,f32) * B(4x16,f32) + C(16x16,f32)
    return __builtin_amdgcn_wmma_f32_16x16x4_f32(
        false, a, false, b, (short)0, c, false, false);
}

// Branchless tanh: no exec-mask divergence, uses v_exp_f32 + v_rcp_f32.
static __device__ __forceinline__ float fast_tanh(float x) {
    const float ax = __builtin_fabsf(x);
    const float t  = __expf(-2.0f * ax);               // in (0,1]
    const float r  = (1.0f - t) * __builtin_amdgcn_rcpf(1.0f + t);
    return __builtin_copysignf(r, x);
}

// ---------------------------------------------------------------------------
// Workspace zero-fill (p1_agg: 25000*128, p3_agg: 25000*192 floats = 32MB)
// ---------------------------------------------------------------------------
__global__ void zero_ws_kernel(float4* ws, long n4) {
    long i = (long)blockIdx.x * blockDim.x + threadIdx.x;
    long stride = (long)gridDim.x * blockDim.x;
    float4 z = {0.f, 0.f, 0.f, 0.f};
    for (; i < n4; i += stride) ws[i] = z;
}

// ---------------------------------------------------------------------------
// Pair kernel: one wave32 per 16-pair tile (grid-stride over tiles)
// 256 threads = 8 waves/WGP (2 per SIMD32) with one 192KB-LDS workgroup.
// ---------------------------------------------------------------------------
__global__ void pair_kernel(const int2* __restrict__ ind2,
                            const float* __restrict__ p1,
                            const float* __restrict__ p3,
                            const float* __restrict__ d3,
                            const float* __restrict__ basis,
                            const float* __restrict__ W_pi,
                            const float* __restrict__ b_pi,
                            const float* __restrict__ W_ii,
                            const float* __restrict__ wi_pix,
                            const float* __restrict__ wj_pix,
                            float* __restrict__ out_i1,
                            float* __restrict__ out_ix,
                            float* __restrict__ agg1,
                            float* __restrict__ agg3) {
    extern __shared__ float lds[];
    const int tid  = threadIdx.x;
    const int lane = tid & 31;
    const int wave = tid >> 5;
    const int half = lane >> 4;     // 0: lanes 0-15, 1: lanes 16-31
    const int l16  = lane & 15;
    const int l8   = lane & 7;
    const int cl   = (lane >> 3) & 1;

    // ---- Stage weights into LDS in (k,k+1)-paired layout: one ds_load_b64
    //      yields the B fragment {W[k][n], W[k+1][n]}.
    for (int f = tid; f < 128 * 256; f += blockDim.x) {      // W_pi 128x256
        int k = f >> 8, n = f & 255;
        lds[WPI_OFF + (k >> 1) * 512 + n * 2 + (k & 1)] = W_pi[f];
    }
    for (int f = tid; f < 32 * 128; f += blockDim.x) {       // W_ii 32x128
        int k = f >> 7, n = f & 127;
        lds[WII_OFF + (k >> 1) * 256 + n * 2 + (k & 1)] = W_ii[f];
    }
    for (int f = tid; f < 64 * 64; f += blockDim.x) {        // wi_pix 64x64
        int k = f >> 6, n = f & 63;
        lds[WIX_OFF + (k >> 1) * 128 + n * 2 + (k & 1)] = wi_pix[f];
    }
    for (int f = tid; f < 64 * 64; f += blockDim.x) {        // wj_pix 64x64
        int k = f >> 6, n = f & 63;
        lds[WJX_OFF + (k >> 1) * 128 + n * 2 + (k & 1)] = wj_pix[f];
    }
    __syncthreads();

    float* i1pre = lds + I1P_OFF + wave * (16 * 32);   // per-wave 16x32 buffer

    // bias is tile-invariant: hoist out of the tile loop
    float biasv[16];
    #pragma unroll
    for (int nt = 0; nt < 16; ++nt) biasv[nt] = b_pi[nt * 16 + l16];

    const int waveGlobal = blockIdx.x * (blockDim.x >> 5) + wave;
    const int waveStride = gridDim.x * (blockDim.x >> 5);

    for (int tile = waveGlobal; tile < N_TILES; tile += waveStride) {
        const int base = tile * 16;
        // A-row indices for the A-fragment rows (m = l16)
        const int2 idxm = ind2[base + l16];
        // row indices / basis / d3 for the C-layout rows this lane owns
        int2  idxr[8];
        float basv[8];
        float d3v[24];
        #pragma unroll
        for (int r = 0; r < 8; ++r) {
            const int m = base + r + 8 * half;
            idxr[r] = ind2[m];
            basv[r] = basis[m * 8 + l8];
            const float3 dv = ((const float3*)d3)[m];
            d3v[0 * 8 + r] = dv.x;
            d3v[1 * 8 + r] = dv.y;
            d3v[2 * 8 + r] = dv.z;
        }

        // hoisted per-lane A base pointers (immediate-offset loads inside)
        const v2f* aI1 = (const v2f*)(p1 + (size_t)idxm.x * 64 + half * 2);
        const v2f* aJ1 = (const v2f*)(p1 + (size_t)idxm.y * 64 + half * 2);

        // ================= GEMM1: [16,128] x W_pi[128,256], tanh, basis =====
        for (int nt = 0; nt < 16; nt += 4) {
            v8f acc[4] = {{}, {}, {}, {}};
            const int ntoff = (nt * 16 + l16) * 2;
            // ---- K = 0..63 (ii half), software-pipelined one K-step ahead
            {
                const float* bb = lds + WPI_OFF + half * 512 + ntoff;
                v2f a_c  = aI1[0];
                v2f b0_c = *(const v2f*)(bb +  0);
                v2f b1_c = *(const v2f*)(bb + 32);
                v2f b2_c = *(const v2f*)(bb + 64);
                v2f b3_c = *(const v2f*)(bb + 96);
                #pragma unroll 5
                for (int kt = 0; kt < 15; ++kt) {
                    const float* bn = bb + (kt + 1) * 1024;
                    const v2f a_n  = aI1[(kt + 1) * 2];
                    const v2f b0_n = *(const v2f*)(bn +  0);
                    const v2f b1_n = *(const v2f*)(bn + 32);
                    const v2f b2_n = *(const v2f*)(bn + 64);
                    const v2f b3_n = *(const v2f*)(bn + 96);
                    acc[0] = wmma4(a_c, b0_c, acc[0]);
                    acc[1] = wmma4(a_c, b1_c, acc[1]);
                    acc[2] = wmma4(a_c, b2_c, acc[2]);
                    acc[3] = wmma4(a_c, b3_c, acc[3]);
                    a_c = a_n; b0_c = b0_n; b1_c = b1_n; b2_c = b2_n; b3_c = b3_n;
                }
                acc[0] = wmma4(a_c, b0_c, acc[0]);
                acc[1] = wmma4(a_c, b1_c, acc[1]);
                acc[2] = wmma4(a_c, b2_c, acc[2]);
                acc[3] = wmma4(a_c, b3_c, acc[3]);
            }
            // ---- K = 64..127 (jj half)
            {
                const float* bb = lds + WPI_OFF + (32 + half) * 512 + ntoff;
                v2f a_c  = aJ1[0];
                v2f b0_c = *(const v2f*)(bb +  0);
                v2f b1_c = *(const v2f*)(bb + 32);
                v2f b2_c = *(const v2f*)(bb + 64);
                v2f b3_c = *(const v2f*)(bb + 96);
                #pragma unroll 5
                for (int kt = 0; kt < 15; ++kt) {
                    const float* bn = bb + (kt + 1) * 1024;
                    const v2f a_n  = aJ1[(kt + 1) * 2];
                    const v2f b0_n = *(const v2f*)(bn +  0);
                    const v2f b1_n = *(const v2f*)(bn + 32);
                    const v2f b2_n = *(const v2f*)(bn + 64);
                    const v2f b3_n = *(const v2f*)(bn + 96);
                    acc[0] = wmma4(a_c, b0_c, acc[0]);
                    acc[1] = wmma4(a_c, b1_c, acc[1]);
                    acc[2] = wmma4(a_c, b2_c, acc[2]);
                    acc[3] = wmma4(a_c, b3_c, acc[3]);
                    a_c = a_n; b0_c = b0_n; b1_c = b1_n; b2_c = b2_n; b3_c = b3_n;
                }
                acc[0] = wmma4(a_c, b0_c, acc[0]);
                acc[1] = wmma4(a_c, b1_c, acc[1]);
                acc[2] = wmma4(a_c, b2_c, acc[2]);
                acc[3] = wmma4(a_c, b3_c, acc[3]);
            }
            #pragma unroll
            for (int u = 0; u < 4; ++u) {
                const int ntc = nt + u;
                const float bias = biasv[ntc];
                #pragma unroll
                for (int r = 0; r < 8; ++r) {
                    float it = fast_tanh(acc[u][r] + bias);
                    float s  = it * basv[r];          // basis[m][b], b = l8
                    s += __shfl_xor(s, 1);
                    s += __shfl_xor(s, 2);
                    s += __shfl_xor(s, 4);            // reduce over b in-group
                    if (l8 == 0)
                        i1pre[(r + 8 * half) * 32 + (ntc * 2 + cl)] = s;
                }
            }
        }

        // ================= GEMM2: i1_pre[16,32] x W_ii[32,128], tanh ========
        float i1keep[4][8];   // columns 64..127 (i1_1) kept for stage 3
        const float* aP = i1pre + l16 * 32 + half * 2;
        for (int nt2 = 0; nt2 < 8; nt2 += 4) {
            v8f acc[4] = {{}, {}, {}, {}};
            const float* bb = lds + WII_OFF + half * 256 + (nt2 * 16 + l16) * 2;
            #pragma unroll
            for (int kt = 0; kt < 8; ++kt) {
                const v2f a = *(const v2f*)(aP + kt * 4);
                const float* bk = bb + kt * 512;
                const v2f b0 = *(const v2f*)(bk +  0);
                const v2f b1 = *(const v2f*)(bk + 32);
                const v2f b2 = *(const v2f*)(bk + 64);
                const v2f b3 = *(const v2f*)(bk + 96);
                acc[0] = wmma4(a, b0, acc[0]);
                acc[1] = wmma4(a, b1, acc[1]);
                acc[2] = wmma4(a, b2, acc[2]);
                acc[3] = wmma4(a, b3, acc[3]);
            }
            #pragma unroll
            for (int u = 0; u < 4; ++u) {
                const int ntc = nt2 + u;
                #pragma unroll
                for (int r = 0; r < 8; ++r) {
                    const float val = fast_tanh(acc[u][r]);
                    const int n = ntc * 16 + l16;
                    const int m = base + r + 8 * half;
                    out_i1[(size_t)m * 128 + n] = val;
                    atomicAdd(&agg1[(size_t)idxr[r].x * 128 + n], val);
                    if (ntc >= 4) i1keep[ntc - 4][r] = val;
                }
            }
        }

        // ===== GEMM3: (p3[ii]@wi_pix + p3[jj]@wj_pix + d3) * i1_1, per x ====
        for (int x = 0; x < 3; ++x) {
            const v2f* aI3 =
                (const v2f*)(p3 + ((size_t)idxm.x * 3 + x) * 64 + half * 2);
            const v2f* aJ3 =
                (const v2f*)(p3 + ((size_t)idxm.y * 3 + x) * 64 + half * 2);
            const float* bi = lds + WIX_OFF + half * 128 + l16 * 2;
            const float* bj = lds + WJX_OFF + half * 128 + l16 * 2;
            v8f acc[4] = {{}, {}, {}, {}};
            // software-pipelined: 10 rotation regs (2 A + 8 B fragments)
            v2f ai_c  = aI3[0],                  aj_c  = aJ3[0];
            v2f bi0_c = *(const v2f*)(bi +  0),  bj0_c = *(const v2f*)(bj +  0);
            v2f bi1_c = *(const v2f*)(bi + 32),  bj1_c = *(const v2f*)(bj + 32);
            v2f bi2_c = *(const v2f*)(bi + 64),  bj2_c = *(const v2f*)(bj + 64);
            v2f bi3_c = *(const v2f*)(bi + 96),  bj3_c = *(const v2f*)(bj + 96);
            #pragma unroll 3
            for (int kt = 0; kt < 15; ++kt) {
                const float* bin = bi + (kt + 1) * 256;
                const float* bjn = bj + (kt + 1) * 256;
                const v2f ai_n  = aI3[(kt + 1) * 2];
                const v2f aj_n  = aJ3[(kt + 1) * 2];
                const v2f bi0_n = *(const v2f*)(bin +  0);
                const v2f bi1_n = *(const v2f*)(bin + 32);
                const v2f bi2_n = *(const v2f*)(bin + 64);
                const v2f bi3_n = *(const v2f*)(bin + 96);
                const v2f bj0_n = *(const v2f*)(bjn +  0);
                const v2f bj1_n = *(const v2f*)(bjn + 32);
                const v2f bj2_n = *(const v2f*)(bjn + 64);
                const v2f bj3_n = *(const v2f*)(bjn + 96);
                acc[0] = wmma4(ai_c, bi0_c, acc[0]);
                acc[1] = wmma4(ai_c, bi1_c, acc[1]);
                acc[2] = wmma4(ai_c, bi2_c, acc[2]);
                acc[3] = wmma4(ai_c, bi3_c, acc[3]);
                acc[0] = wmma4(aj_c, bj0_c, acc[0]);
                acc[1] = wmma4(aj_c, bj1_c, acc[1]);
                acc[2] = wmma4(aj_c, bj2_c, acc[2]);
                acc[3] = wmma4(aj_c, bj3_c, acc[3]);
                ai_c = ai_n;   aj_c = aj_n;
                bi0_c = bi0_n; bi1_c = bi1_n; bi2_c = bi2_n; bi3_c = bi3_n;
                bj0_c = bj0_n; bj1_c = bj1_n; bj2_c = bj2_n; bj3_c = bj3_n;
            }
            acc[0] = wmma4(ai_c, bi0_c, acc[0]);
            acc[1] = wmma4(ai_c, bi1_c, acc[1]);
            acc[2] = wmma4(ai_c, bi2_c, acc[2]);
            acc[3] = wmma4(ai_c, bi3_c, acc[3]);
            acc[0] = wmma4(aj_c, bj0_c, acc[0]);
            acc[1] = wmma4(aj_c, bj1_c, acc[1]);
            acc[2] = wmma4(aj_c, bj2_c, acc[2]);
            acc[3] = wmma4(aj_c, bj3_c, acc[3]);

            #pragma unroll
            for (int u = 0; u < 4; ++u) {
                #pragma unroll
                for (int r = 0; r < 8; ++r) {
                    const int n = u * 16 + l16;
                    const float val =
                        (acc[u][r] + d3v[x * 8 + r]) * i1keep[u][r];
                    const int m = base + r + 8 * half;
                    out_ix[((size_t)m * 3 + x) * 64 + n] = val;
                    atomicAdd(&agg3[((size_t)idxr[r].x * 3 + x) * 64 + n], val);
                }
            }
        }
    }
}

// ---------------------------------------------------------------------------
// Atom kernel: one 128-thread block per atom (small GEMM tail, VALU)
// ---------------------------------------------------------------------------
__global__ void atom_kernel(const float* __restrict__ agg1,
                            const float* __restrict__ agg3,
                            const float* __restrict__ W_pp,
                            const float* __restrict__ W_ppx,
                            const float* __restrict__ wi_dot,
                            const float* __restrict__ wj_dot,
                            const float* __restrict__ W_pp1,
                            const float* __restrict__ b_pp1,
                            float* __restrict__ out_p1,
                            float* __restrict__ out_p3,
                            float* __restrict__ out_dot) {
    __shared__ float s_p3m[192];
    __shared__ float s_cat[128];
    __shared__ float s_s3[64];
    const int a = blockIdx.x;
    const int t = threadIdx.x;
    const float* g1 = agg1 + (size_t)a * 128;
    const float* g3 = agg3 + (size_t)a * 192;

    if (t < 64) {                       // p1_new = tanh(p1_agg @ W_pp)
        float acc = 0.f;
        for (int k = 0; k < 128; ++k) acc += g1[k] * W_pp[k * 64 + t];
        s_cat[t] = fast_tanh(acc);
    }
    for (int idx = t; idx < 192; idx += 128) {   // p3_mixed = p3_agg @ W_ppx
        const int x = idx >> 6, d = idx & 63;
        float acc = 0.f;
        for (int c = 0; c < 64; ++c) acc += g3[x * 64 + c] * W_ppx[c * 64 + d];
        s_p3m[idx] = acc;
    }
    __syncthreads();
    if (t < 64) {                       // dotted_p3
        float acc = 0.f;
        for (int x = 0; x < 3; ++x) {
            float s1 = 0.f, s2 = 0.f;
            for (int c = 0; c < 64; ++c) {
                const float pv = s_p3m[x * 64 + c];
                s1 += pv * wi_dot[c * 64 + t];
                s2 += pv * wj_dot[c * 64 + t];
            }
            acc += s1 * s2;
        }
        s_cat[64 + t] = acc;
        out_dot[(size_t)a * 64 + t] = acc;
    }
    __syncthreads();
    {                                   // p1t1 = tanh(cat @ W_pp1 + b)
        float acc = b_pp1[t];
        for (int k = 0; k < 128; ++k) acc += s_cat[k] * W_pp1[k * 128 + t];
        const float v = fast_tanh(acc);
        if (t < 64) out_p1[(size_t)a * 64 + t] = v;
        else        s_s3[t - 64] = v;
    }
    __syncthreads();
    for (int idx = t; idx < 192; idx += 128)
        out_p3[(size_t)a * 192 + idx] = s_p3m[idx] * s_s3[idx & 63];
}

// ---------------------------------------------------------------------------
extern "C" void kernel_launch(void* const* d_in, const int* in_sizes, int n_in,
                              void* d_out, int out_size, void* d_ws, size_t ws_size,
                              hipStream_t stream) {
    const int2*  ind2   = (const int2*) d_in[0];
    const float* p1     = (const float*)d_in[1];
    const float* p3     = (const float*)d_in[2];
    const float* d3     = (const float*)d_in[3];
    const float* basis  = (const float*)d_in[4];
    const float* W_pi   = (const float*)d_in[5];
    const float* b_pi   = (const float*)d_in[6];
    const float* W_ii   = (const float*)d_in[7];
    const float* W_pp   = (const float*)d_in[8];
    const float* wi_pix = (const float*)d_in[9];
    const float* wj_pix = (const float*)d_in[10];
    const float* W_ppx  = (const float*)d_in[11];
    const float* wi_dot = (const float*)d_in[12];
    const float* wj_dot = (const float*)d_in[13];
    const float* W_pp1  = (const float*)d_in[14];
    const float* b_pp1  = (const float*)d_in[15];

    float* out = (float*)d_out;
    float* out_p1  = out;                                   // 25000*64
    float* out_p3  = out + (size_t)N_ATOMS_C * 64;          // 25000*192
    float* out_dot = out + (size_t)N_ATOMS_C * (64 + 192);  // 25000*64
    float* out_i1  = out + (size_t)N_ATOMS_C * (64 + 192 + 64);       // 500000*128
    float* out_ix  = out_i1 + (size_t)N_PAIRS_C * 128;      // 500000*192

    float* agg1 = (float*)d_ws;                       // 25000*128
    float* agg3 = agg1 + (size_t)N_ATOMS_C * 128;     // 25000*192

    // zero the segment-sum accumulators (32 MB) every call
    const long n4 = (long)N_ATOMS_C * (128 + 192) / 4;
    zero_ws_kernel<<<2048, 256, 0, stream>>>((float4*)d_ws, n4);

    hipFuncSetAttribute((const void*)pair_kernel,
                        hipFuncAttributeMaxDynamicSharedMemorySize, LDS_BYTES);
    pair_kernel<<<384, 256, LDS_BYTES, stream>>>(
        ind2, p1, p3, d3, basis, W_pi, b_pi, W_ii, wi_pix, wj_pix,
        out_i1, out_ix, agg1, agg3);

    atom_kernel<<<N_ATOMS_C, 128, 0, stream>>>(
        agg1, agg3, W_pp, W_ppx, wi_dot, wj_dot, W_pp1, b_pp1,
        out_p1, out_p3, out_dot);
}